// RadiusSearch_22119081574483
// MI455X (gfx1250) — compile-verified
//
#include <hip/hip_runtime.h>

typedef __attribute__((ext_vector_type(2))) float v2f;
typedef __attribute__((ext_vector_type(4))) float v4f;
typedef __attribute__((ext_vector_type(8))) float v8f;

#define TILE 16
#define NCHUNKS 16
#define WAVES_PER_BLOCK 8

// ---------------------------------------------------------------------------
// Pass 0: fill padding. index region = -1, splits + distance regions = 0.
// 268 MB of writes -> the bandwidth floor of the whole problem. Use B128
// stores so the fill is bandwidth- (not issue-) limited.  mn is 16-aligned
// here so each float4 sits entirely in one region.
// ---------------------------------------------------------------------------
__global__ void rs_init(float* __restrict__ out, long long mn, long long total) {
  const long long n4 = total >> 2;
  long long i = (long long)blockIdx.x * blockDim.x + threadIdx.x;
  const long long stride = (long long)gridDim.x * blockDim.x;
  v4f* __restrict__ out4 = (v4f*)out;
  for (long long j = i; j < n4; j += stride) {
    long long e = j << 2;               // first element of this float4
    float v = (e < mn) ? -1.0f : 0.0f;  // mn % 4 == 0 -> uniform within vector
    v4f w = {v, v, v, v};
    out4[j] = w;
  }
  if (i == 0)
    for (long long e = n4 << 2; e < total; ++e) out[e] = 0.0f;  // tail (end of distance region)
}

// ---------------------------------------------------------------------------
// Per-wave WMMA tile setup for a 16-query tile.
// A (16x4 f32): lanes 0-15 hold (qx,qy) for M=lane&15; lanes 16-31 hold (qz,1).
// C (16x16 f32): cq[r] = |q|^2 of row tileM + r + (lane>=16 ? 8 : 0).
// ---------------------------------------------------------------------------
__device__ __forceinline__ void rs_setup_tile(const float* __restrict__ qrs,
                                              const float* __restrict__ rad,
                                              int tileM, int lane,
                                              v2f& a, v8f& cq, float r2[8]) {
  const int hi = (lane >= 16) ? 8 : 0;
  const int mA = tileM + (lane & 15);
  float qx = qrs[3 * mA + 0];
  float qy = qrs[3 * mA + 1];
  float qz = qrs[3 * mA + 2];
  a.x = (lane < 16) ? qx : qz;
  a.y = (lane < 16) ? qy : 1.0f;
#pragma unroll
  for (int r = 0; r < 8; ++r) {
    int row = tileM + r + hi;
    float x = qrs[3 * row + 0];
    float y = qrs[3 * row + 1];
    float z = qrs[3 * row + 2];
    cq[r] = x * x + y * y + z * z;
    float rr = rad[row];
    r2[r] = rr * rr;
  }
}

// B (4x16 f32) for a 16-point tile starting at n0:
// lanes 0-15 hold (B[0][n],B[1][n]) = (-2px,-2py); lanes 16-31 hold (-2pz,|p|^2).
__device__ __forceinline__ v2f rs_load_b(const float* __restrict__ pts, int n0,
                                         int lane) {
  const int n = n0 + (lane & 15);
  float px = pts[3 * n + 0];
  float py = pts[3 * n + 1];
  float pz = pts[3 * n + 2];
  v2f b;
  b.x = (lane < 16) ? (-2.0f * px) : (-2.0f * pz);
  b.y = (lane < 16) ? (-2.0f * py) : (px * px + py * py + pz * pz);
  return b;
}

// ---------------------------------------------------------------------------
// Pass 1: per-(query-tile, chunk) neighbor counts via WMMA + wave32 ballot.
// One wave owns 16 queries x (N/NCHUNKS) points.  Note: r2 > 0, so the
// reference's max(d2,0) <= r2 is equivalent to d2 <= r2 -> no clamp needed.
// ---------------------------------------------------------------------------
__global__ __launch_bounds__(256) void rs_count(const float* __restrict__ pts,
                                                const float* __restrict__ qrs,
                                                const float* __restrict__ rad,
                                                unsigned* __restrict__ counts,
                                                int M, int N) {
  const int lane = threadIdx.x & 31;
  const int wave = blockIdx.x * WAVES_PER_BLOCK + (threadIdx.x >> 5);
  const int tile = wave / NCHUNKS;
  const int chunk = wave - tile * NCHUNKS;
  const int tileM = tile * TILE;
  const int chunkLen = N / NCHUNKS;
  const int n_base = chunk * chunkLen;

  v2f a; v8f cq; float r2[8];
  rs_setup_tile(qrs, rad, tileM, lane, a, cq, r2);

  unsigned cnt[16];
#pragma unroll
  for (int i = 0; i < 16; ++i) cnt[i] = 0;

#pragma unroll 2
  for (int t = 0; t < chunkLen; t += TILE) {
    __builtin_prefetch(&pts[3 * (n_base + t + TILE)], 0, 3);
    v2f b = rs_load_b(pts, n_base + t, lane);
    v8f d = __builtin_amdgcn_wmma_f32_16x16x4_f32(
        false, a, false, b, (short)0, cq, false, false);
#pragma unroll
    for (int r = 0; r < 8; ++r) {
      bool pass = d[r] <= r2[r];
      unsigned bal = (unsigned)__ballot(pass);  // [15:0]=rows r, [31:16]=rows r+8
      cnt[r]     += __popc(bal & 0xFFFFu);
      cnt[r + 8] += __popc(bal >> 16);
    }
  }
  if (lane < 16) counts[(size_t)(tileM + lane) * NCHUNKS + chunk] = cnt[lane];
}

// ---------------------------------------------------------------------------
// Pass 2: single-block scan -> row_splits (into d_out) and per-(row,chunk)
// packed-write start offsets (into workspace).
// ---------------------------------------------------------------------------
__global__ __launch_bounds__(256) void rs_scan(const unsigned* __restrict__ counts,
                                               unsigned* __restrict__ offs,
                                               float* __restrict__ splits, int M) {
  __shared__ unsigned sh[256];
  const int t = threadIdx.x;
  const int rpt = M / 256;

  unsigned sum = 0;
  for (int i = 0; i < rpt; ++i) {
    int row = t * rpt + i;
    for (int c = 0; c < NCHUNKS; ++c) sum += counts[(size_t)row * NCHUNKS + c];
  }
  sh[t] = sum;
  __syncthreads();
  for (int o = 1; o < 256; o <<= 1) {
    unsigned v = (t >= o) ? sh[t - o] : 0u;
    __syncthreads();
    sh[t] += v;
    __syncthreads();
  }
  unsigned run = (t > 0) ? sh[t - 1] : 0u;
  for (int i = 0; i < rpt; ++i) {
    int row = t * rpt + i;
    splits[row] = (float)run;
    for (int c = 0; c < NCHUNKS; ++c) {
      offs[(size_t)row * NCHUNKS + c] = run;
      run += counts[(size_t)row * NCHUNKS + c];
    }
  }
  if (t == 255) splits[M] = (float)run;
}

// ---------------------------------------------------------------------------
// Pass 3: recompute tiles via WMMA, pack neighbors deterministically.
// Ballot masks are wave-uniform, so all lanes track all 16 row offsets in
// registers; per-lane write position = offset[row] + rank-within-half.
// Query-major, point-index order preserved -> matches reference packing.
// ---------------------------------------------------------------------------
__global__ __launch_bounds__(256) void rs_pack(const float* __restrict__ pts,
                                               const float* __restrict__ qrs,
                                               const float* __restrict__ rad,
                                               const unsigned* __restrict__ offs,
                                               float* __restrict__ idx_out,
                                               float* __restrict__ dist_out,
                                               int M, int N) {
  const int lane = threadIdx.x & 31;
  const int wave = blockIdx.x * WAVES_PER_BLOCK + (threadIdx.x >> 5);
  const int tile = wave / NCHUNKS;
  const int chunk = wave - tile * NCHUNKS;
  const int tileM = tile * TILE;
  const int chunkLen = N / NCHUNKS;
  const int n_base = chunk * chunkLen;
  const int hi = (lane >= 16) ? 8 : 0;

  v2f a; v8f cq; float r2[8];
  rs_setup_tile(qrs, rad, tileM, lane, a, cq, r2);

  unsigned off[16];
#pragma unroll
  for (int i = 0; i < 16; ++i)
    off[i] = offs[(size_t)(tileM + i) * NCHUNKS + chunk];

#pragma unroll 2
  for (int t = 0; t < chunkLen; t += TILE) {
    const int n0 = n_base + t;
    __builtin_prefetch(&pts[3 * (n0 + TILE)], 0, 3);
    v2f b = rs_load_b(pts, n0, lane);
    v8f d = __builtin_amdgcn_wmma_f32_16x16x4_f32(
        false, a, false, b, (short)0, cq, false, false);
#pragma unroll
    for (int r = 0; r < 8; ++r) {
      float dv = fmaxf(d[r], 0.0f);     // stored value must match reference
      bool pass = dv <= r2[r];
      unsigned bal = (unsigned)__ballot(pass);
      unsigned halfMask = (lane < 16) ? (bal & 0xFFFFu) : (bal >> 16);
      unsigned laneInHalf = lane & 15;
      if (pass) {
        unsigned rank = __popc(halfMask & ((1u << laneInHalf) - 1u));
        unsigned pos = off[r + hi] + rank;
        idx_out[pos] = (float)(n0 + (int)laneInHalf);
        dist_out[pos] = dv;
      }
      off[r]     += __popc(bal & 0xFFFFu);
      off[r + 8] += __popc(bal >> 16);
    }
  }
}

// ---------------------------------------------------------------------------
extern "C" void kernel_launch(void* const* d_in, const int* in_sizes, int n_in,
                              void* d_out, int out_size, void* d_ws, size_t ws_size,
                              hipStream_t stream) {
  const float* pts = (const float*)d_in[0];
  const float* qrs = (const float*)d_in[1];
  const float* rad = (const float*)d_in[2];
  const int N = in_sizes[0] / 3;   // 8192
  const int M = in_sizes[2];       // 4096
  const long long MN = (long long)M * N;
  const long long total = (long long)out_size;  // 2*MN + M + 1

  float* out = (float*)d_out;
  float* idx_out = out;                  // [M*N] indices (as float, -1 pad)
  float* splits_out = out + MN;          // [M+1] row splits
  float* dist_out = out + MN + (M + 1);  // [M*N] distances (0 pad)

  unsigned* counts = (unsigned*)d_ws;                  // [M][NCHUNKS]
  unsigned* offs = counts + (size_t)M * NCHUNKS;       // [M][NCHUNKS]

  rs_init<<<2048, 256, 0, stream>>>(out, MN, total);

  const int tiles = M / TILE;                       // 256
  const int waves = tiles * NCHUNKS;                // 4096
  const int blocks = waves / WAVES_PER_BLOCK;       // 512
  rs_count<<<blocks, 256, 0, stream>>>(pts, qrs, rad, counts, M, N);
  rs_scan<<<1, 256, 0, stream>>>(counts, offs, splits_out, M);
  rs_pack<<<blocks, 256, 0, stream>>>(pts, qrs, rad, offs, idx_out, dist_out, M, N);
}